// LogicalDiscriminator_3143916060807
// MI455X (gfx1250) — compile-verified
//
#include <hip/hip_runtime.h>
#include <hip/hip_bf16.h>
#include <math.h>

#define HID 128
#define HALF_HID 64

typedef __attribute__((ext_vector_type(2))) float v2f;
typedef __attribute__((ext_vector_type(8))) float v8f;

// ---------------------------------------------------------------- fill
__global__ void fill_kernel(float* __restrict__ p, float v, long n) {
    long i = (long)blockIdx.x * blockDim.x + threadIdx.x;
    if (i < n) p[i] = v;
}

// ---------------------------------------------------------------- degree: deg[col] += 1 per edge (deg pre-filled with 1.0 for self-loop)
__global__ void degree_kernel(const int* __restrict__ ei, float* __restrict__ deg, int E) {
    int e = blockIdx.x * blockDim.x + threadIdx.x;
    if (e < E) atomicAdd(&deg[ei[E + e]], 1.0f);   // col = edge_index[1][e]
}

// ---------------------------------------------------------------- dinv = rsqrt(deg)  (deg >= 1 always due to self loop)
__global__ void rsqrt_kernel(const float* __restrict__ deg, float* __restrict__ dinv, int N) {
    int i = blockIdx.x * blockDim.x + threadIdx.x;
    if (i < N) dinv[i] = rsqrtf(deg[i]);
}

// ---------------------------------------------------------------- h = x @ W_emb + b_emb   (rank-1), one wave per node, float4/lane
__global__ void embed_kernel(const float* __restrict__ x, const float* __restrict__ Wemb,
                             const float* __restrict__ bemb, float* __restrict__ h, int N) {
    int gid  = blockIdx.x * blockDim.x + threadIdx.x;
    int node = gid >> 5;
    int lane = gid & 31;
    if (node >= N) return;
    float xv = x[node];
    int j = lane * 4;
    float4 w = *(const float4*)(Wemb + j);
    float4 b = *(const float4*)(bemb + j);
    float4 o;
    o.x = fmaf(xv, w.x, b.x); o.y = fmaf(xv, w.y, b.y);
    o.z = fmaf(xv, w.z, b.z); o.w = fmaf(xv, w.w, b.w);
    *(float4*)(h + (size_t)node * HID + j) = o;
}

// ---------------------------------------------------------------- Hout = Hin @ W  via V_WMMA_F32_16X16X4_F32
// 256 threads = 8 waves; block covers 128 rows (wave -> 16 rows x 128 cols).
// W staged in LDS pair-interleaved: Wlds2[p*128 + n] = {W[2p][n], W[2p+1][n]}
// so every B fragment is one aligned ds_load_b64 straight into an even VGPR pair.
__global__ __launch_bounds__(256) void gemm_wmma_kernel(
        const float* __restrict__ Hin, const float* __restrict__ W,
        float* __restrict__ Hout, int N) {
    __shared__ float Wlds[HID * HID];          // viewed as v2f[64*128]
    v2f* Wp = (v2f*)Wlds;
    {   // cooperative load + 2x2 register transpose into pair-interleaved layout
        for (int i = threadIdx.x; i < HALF_HID * HALF_HID; i += 256) {
            int p  = i >> 6;                   // row pair 0..63
            int n2 = (i & 63) * 2;             // even column
            float2 r0 = *(const float2*)(W + (size_t)(2 * p)     * HID + n2);
            float2 r1 = *(const float2*)(W + (size_t)(2 * p + 1) * HID + n2);
            v2f a; a.x = r0.x; a.y = r1.x;
            v2f b; b.x = r0.y; b.y = r1.y;
            Wp[p * HID + n2]     = a;
            Wp[p * HID + n2 + 1] = b;
        }
    }
    __syncthreads();

    int wave = threadIdx.x >> 5;
    int lane = threadIdx.x & 31;
    int row0 = blockIdx.x * 128 + wave * 16;
    if (row0 >= N) return;                     // N % 16 == 0 -> full tiles only

    int m  = lane & 15;                        // A: lane -> M
    int n  = lane & 15;                        // B/C/D: lane -> N
    int hi = lane >> 4;                        // upper lane-half selector

    v8f acc[8] = {};                           // 8 N-tiles of the 16x128 stripe

    const float* arow = Hin + (size_t)(row0 + m) * HID;
    for (int k0 = 0; k0 < HID; k0 += 4) {
        int ka = k0 + 2 * hi;                  // even -> 8B aligned
        v2f a = *(const v2f*)(arow + ka);      // A: VGPR0=K+0/K+2, VGPR1=K+1/K+3
        int p = ka >> 1;                       // pair index in LDS
        #pragma unroll
        for (int t = 0; t < 8; ++t) {
            v2f b = Wp[p * HID + t * 16 + n];  // {W[ka][col], W[ka+1][col]}
            acc[t] = __builtin_amdgcn_wmma_f32_16x16x4_f32(
                false, a, false, b, (short)0, acc[t], false, false);
        }
    }
    // C/D layout: lane n -> column, VGPR v -> row v + 8*hi
    #pragma unroll
    for (int t = 0; t < 8; ++t) {
        int col = t * 16 + n;
        #pragma unroll
        for (int v = 0; v < 8; ++v)
            Hout[(size_t)(row0 + v + 8 * hi) * HID + col] = acc[t][v];
    }
}

// ---------------------------------------------------------------- per-edge scatter: agg[col] += hlin[row] * dinv[row]*dinv[col]
// one wave per edge, float4 per lane, 4x global_atomic_add_f32
__global__ void scatter_kernel(const int* __restrict__ ei, const float* __restrict__ hlin,
                               const float* __restrict__ dinv, float* __restrict__ agg, int E) {
    long gid = (long)blockIdx.x * blockDim.x + threadIdx.x;
    int e    = (int)(gid >> 5);
    int lane = (int)(gid & 31);
    if (e >= E) return;
    int r = ei[e];          // source
    int c = ei[E + e];      // target
    float nrm = dinv[r] * dinv[c];
    float4 v = ((const float4*)(hlin + (size_t)r * HID))[lane];
    float* dst = agg + (size_t)c * HID + lane * 4;
    atomicAdd(dst + 0, v.x * nrm);
    atomicAdd(dst + 1, v.y * nrm);
    atomicAdd(dst + 2, v.z * nrm);
    atomicAdd(dst + 3, v.w * nrm);
}

// ---------------------------------------------------------------- h = relu(agg + hlin*dinv^2 (self loop) + b)
__global__ void relu_bias_kernel(const float* __restrict__ agg, const float* __restrict__ hlin,
                                 const float* __restrict__ dinv, const float* __restrict__ b,
                                 float* __restrict__ h, int N) {
    int gid  = blockIdx.x * blockDim.x + threadIdx.x;
    int node = gid >> 5;
    int lane = gid & 31;
    if (node >= N) return;
    float sl = dinv[node]; sl *= sl;           // self-loop norm
    int j = lane * 4;
    float4 a  = *(const float4*)(agg  + (size_t)node * HID + j);
    float4 hl = *(const float4*)(hlin + (size_t)node * HID + j);
    float4 bb = *(const float4*)(b + j);
    float4 o;
    o.x = fmaxf(fmaf(hl.x, sl, a.x) + bb.x, 0.0f);
    o.y = fmaxf(fmaf(hl.y, sl, a.y) + bb.y, 0.0f);
    o.z = fmaxf(fmaf(hl.z, sl, a.z) + bb.z, 0.0f);
    o.w = fmaxf(fmaf(hl.w, sl, a.w) + bb.w, 0.0f);
    *(float4*)(h + (size_t)node * HID + j) = o;
}

// ---------------------------------------------------------------- mean-pool accumulation per graph
__global__ void pool_kernel(const float* __restrict__ h, const int* __restrict__ batch,
                            float* __restrict__ sums, float* __restrict__ counts, int N) {
    int gid  = blockIdx.x * blockDim.x + threadIdx.x;
    int node = gid >> 5;
    int lane = gid & 31;
    if (node >= N) return;
    int g = batch[node];
    float4 v = ((const float4*)(h + (size_t)node * HID))[lane];
    float* dst = sums + (size_t)g * HID + lane * 4;
    atomicAdd(dst + 0, v.x);
    atomicAdd(dst + 1, v.y);
    atomicAdd(dst + 2, v.z);
    atomicAdd(dst + 3, v.w);
    if (lane == 0) atomicAdd(&counts[g], 1.0f);
}

// ---------------------------------------------------------------- head: sigmoid(relu(g@Wc1+bc1)@Wc2+bc2), one thread per graph
__global__ void classifier_kernel(const float* __restrict__ sums, const float* __restrict__ counts,
                                  const float* __restrict__ Wc1, const float* __restrict__ bc1,
                                  const float* __restrict__ Wc2, const float* __restrict__ bc2,
                                  float* __restrict__ out, int G) {
    int g = blockIdx.x * blockDim.x + threadIdx.x;
    if (g >= G) return;
    float inv = 1.0f / fmaxf(counts[g], 1.0f);
    float z[HALF_HID];
    #pragma unroll
    for (int k = 0; k < HALF_HID; ++k) z[k] = bc1[k];
    for (int j = 0; j < HID; ++j) {
        float gj = sums[(size_t)g * HID + j] * inv;
        #pragma unroll
        for (int k = 0; k < HALF_HID; ++k)
            z[k] = fmaf(gj, Wc1[j * HALF_HID + k], z[k]);
    }
    float o = bc2[0];
    #pragma unroll
    for (int k = 0; k < HALF_HID; ++k)
        o = fmaf(fmaxf(z[k], 0.0f), Wc2[k], o);
    out[g] = 1.0f / (1.0f + __expf(-o));
}

// ================================================================ launch
extern "C" void kernel_launch(void* const* d_in, const int* in_sizes, int n_in,
                              void* d_out, int out_size, void* d_ws, size_t ws_size,
                              hipStream_t stream) {
    const float* x     = (const float*)d_in[0];
    const int*   ei    = (const int*)  d_in[1];   // [2, E]
    const int*   batch = (const int*)  d_in[2];
    const float* Wemb  = (const float*)d_in[3];
    const float* bemb  = (const float*)d_in[4];
    const float* Wgnn  = (const float*)d_in[5];   // [3,128,128]
    const float* bgnn  = (const float*)d_in[6];   // [3,128]
    const float* Wc1   = (const float*)d_in[7];
    const float* bc1   = (const float*)d_in[8];
    const float* Wc2   = (const float*)d_in[9];
    const float* bc2   = (const float*)d_in[10];
    float* out = (float*)d_out;

    const int N = in_sizes[0];           // 100000
    const int E = in_sizes[1] / 2;       // 3200000
    const int G = out_size;              // 128
    const int DEPTH = 3;

    // workspace carve-up (floats)
    float* ws     = (float*)d_ws;
    float* deg    = ws;                         ws += N;
    float* dinv   = ws;                         ws += N;
    float* h      = ws;                         ws += (size_t)N * HID;
    float* hlin   = ws;                         ws += (size_t)N * HID;
    float* agg    = ws;                         ws += (size_t)N * HID;
    float* sums   = ws;                         ws += (size_t)G * HID;
    float* counts = ws;                         ws += G;
    (void)ws_size; (void)n_in;

    const int TB = 256;
    const unsigned gN     = (unsigned)((N + TB - 1) / TB);
    const unsigned gE     = (unsigned)((E + TB - 1) / TB);
    const unsigned gNwave = (unsigned)(((long)N * 32 + TB - 1) / TB);
    const unsigned gEwave = (unsigned)(((long)E * 32 + TB - 1) / TB);
    const unsigned gNH    = (unsigned)(((long)N * HID + TB - 1) / TB);
    const unsigned gGemm  = (unsigned)((N + 127) / 128);

    // --- degree & normalization ---
    fill_kernel<<<gN, TB, 0, stream>>>(deg, 1.0f, N);            // self-loop
    degree_kernel<<<gE, TB, 0, stream>>>(ei, deg, E);
    rsqrt_kernel<<<gN, TB, 0, stream>>>(deg, dinv, N);

    // --- embedding ---
    embed_kernel<<<gNwave, TB, 0, stream>>>(x, Wemb, bemb, h, N);

    // --- GCN layers ---
    for (int l = 0; l < DEPTH; ++l) {
        gemm_wmma_kernel<<<gGemm, TB, 0, stream>>>(h, Wgnn + (size_t)l * HID * HID, hlin, N);
        fill_kernel<<<gNH, TB, 0, stream>>>(agg, 0.0f, (long)N * HID);
        scatter_kernel<<<gEwave, TB, 0, stream>>>(ei, hlin, dinv, agg, E);
        relu_bias_kernel<<<gNwave, TB, 0, stream>>>(agg, hlin, dinv, bgnn + (size_t)l * HID, h, N);
    }

    // --- pooling + classifier head ---
    fill_kernel<<<(unsigned)((G * HID + TB - 1) / TB), TB, 0, stream>>>(sums, 0.0f, (long)G * HID);
    fill_kernel<<<1, TB, 0, stream>>>(counts, 0.0f, G);
    pool_kernel<<<gNwave, TB, 0, stream>>>(h, batch, sums, counts, N);
    classifier_kernel<<<(unsigned)((G + 127) / 128), 128, 0, stream>>>(
        sums, counts, Wc1, bc1, Wc2, bc2, out, G);
}